// QuantumNATPlus_65481071407580
// MI455X (gfx1250) — compile-verified
//
#include <hip/hip_runtime.h>

typedef float v2f __attribute__((ext_vector_type(2)));
typedef float v8f __attribute__((ext_vector_type(8)));

// ---------------------------------------------------------------------------
// Kernel 0: zero the 8 batchnorm accumulators (sum, sumsq per channel)
// ---------------------------------------------------------------------------
__global__ void qnat_init(float* __restrict__ accum) {
    if (threadIdx.x < 8) accum[threadIdx.x] = 0.0f;
}

// ---------------------------------------------------------------------------
// Kernel 1: fused pooling + analytic 4-qubit circuit + MLP (WMMA f32) +
//           logits to workspace + batchnorm partial sums (atomics)
// Block = 128 threads = 4 waves; each wave owns a 16-row batch tile.
// ---------------------------------------------------------------------------
__global__ __launch_bounds__(128) void qnat_main(
    const float* __restrict__ x,       // (B,1,36,36)
    const float* __restrict__ qp,      // (8,)
    const float* __restrict__ W1,      // (64,4) row-major
    const float* __restrict__ b1,      // (64,)
    const float* __restrict__ W2,      // (4,64) row-major
    const float* __restrict__ b2,      // (4,)
    float* __restrict__ logits_ws,     // (B,4)
    float* __restrict__ accum)         // (8,) [n*2]=sum, [n*2+1]=sumsq
{
    __shared__ float Hs[4][16 * 68];   // per-wave H tile, padded stride 68
    __shared__ float red[8];           // block-level batchnorm partials

    const int tid  = threadIdx.x;
    const int wave = tid >> 5;
    const int lane = tid & 31;
    const int half = lane >> 4;        // 0 | 1
    const int ln   = lane & 15;        // 0..15

    if (tid < 8) red[tid] = 0.0f;
    __syncthreads();

    const int m0 = blockIdx.x * 64 + wave * 16;   // batch row base of this wave
    const int m  = m0 + ln;                       // this lane's A-matrix row

    // ---- pooling + circuit: lane computes z_k for k = 2*half+v, v=0,1 ----
    // feature_k = mean of x[m, 0, 0:6, 6k:6k+6];  z_k = cos(qp[k])*cos(feature_k)
    float zv[2];
#pragma unroll
    for (int v = 0; v < 2; ++v) {
        const int k = 2 * half + v;
        const float* base = x + (size_t)m * 1296 + (size_t)k * 6;
        float s = 0.0f;
#pragma unroll
        for (int r = 0; r < 6; ++r)
#pragma unroll
            for (int c = 0; c < 6; ++c)
                s += base[r * 36 + c];
        const float feat = s * (1.0f / 36.0f);
        zv[v] = __cosf(qp[k]) * __cosf(feat);
    }

    // assemble z0..z3 per row via half-swap, then prefix products q_k
    const float o0 = __shfl_xor(zv[0], 16, 32);
    const float o1 = __shfl_xor(zv[1], 16, 32);
    const float z0 = half ? o0    : zv[0];
    const float z1 = half ? o1    : zv[1];
    const float z2 = half ? zv[0] : o0;
    const float z3 = half ? zv[1] : o1;
    const float p0 = z0;
    const float p1 = p0 * z1;
    const float p2 = p1 * z2;
    const float p3 = p2 * z3;

    // A-matrix (16x4 f32): lane L, VGPR v -> A[m = L%16][k = 2*(L/16)+v]
    v2f A;
    A.x = half ? p2 : p0;
    A.y = half ? p3 : p1;

    // ---- GEMM1: H(16x64) = relu(A(16x4) x W1^T(4x64) + b1), via 4 WMMAs ----
    float* hrow = &Hs[wave][0];
#pragma unroll
    for (int t = 0; t < 4; ++t) {
        const int n16 = 16 * t + ln;                  // hidden unit index
        v2f Bm;                                       // B[k][n] = W1[n][k]
        Bm.x = W1[n16 * 4 + (2 * half + 0)];
        Bm.y = W1[n16 * 4 + (2 * half + 1)];
        v8f C;
        const float bias = b1[n16];
#pragma unroll
        for (int r = 0; r < 8; ++r) C[r] = bias;
        v8f D = __builtin_amdgcn_wmma_f32_16x16x4_f32(
            false, A, false, Bm, (short)0, C, false, false);
        // D layout: D[m = r + 8*half][n = ln]; relu and stage to LDS
#pragma unroll
        for (int r = 0; r < 8; ++r)
            hrow[(r + 8 * half) * 68 + n16] = fmaxf(D[r], 0.0f);
    }

    // ---- GEMM2: logits(16x4) = H(16x64) x W2^T(64x4, padded) + b2 ----
    v8f acc;
#pragma unroll
    for (int r = 0; r < 8; ++r) acc[r] = (ln < 4) ? b2[ln] : 0.0f;
#pragma unroll
    for (int c = 0; c < 16; ++c) {
        const int k0 = 4 * c + 2 * half;
        v2f A2, B2;
        A2.x = hrow[ln * 68 + k0 + 0];                // A2[m=ln][k]
        A2.y = hrow[ln * 68 + k0 + 1];
        B2.x = (ln < 4) ? W2[ln * 64 + k0 + 0] : 0.0f; // B2[k][n] = W2[n][k]
        B2.y = (ln < 4) ? W2[ln * 64 + k0 + 1] : 0.0f;
        acc = __builtin_amdgcn_wmma_f32_16x16x4_f32(
            false, A2, false, B2, (short)0, acc, false, false);
    }

    // ---- store logits + batchnorm partials ----
    float s = 0.0f, ss = 0.0f;
    if (ln < 4) {
#pragma unroll
        for (int r = 0; r < 8; ++r) {
            const float v = acc[r];                   // logits[m0+r+8*half][ln]
            logits_ws[(size_t)(m0 + r + 8 * half) * 4 + ln] = v;
            s  += v;
            ss += v * v;
        }
    }
    s  += __shfl_down(s, 16, 32);                     // fold upper half into lower
    ss += __shfl_down(ss, 16, 32);
    if (half == 0 && ln < 4) {
        atomicAdd(&red[ln * 2 + 0], s);
        atomicAdd(&red[ln * 2 + 1], ss);
    }
    __syncthreads();
    if (tid < 8) atomicAdd(&accum[tid], red[tid]);
}

// ---------------------------------------------------------------------------
// Kernel 2: batchnorm finalize (elementwise over B*4)
// ---------------------------------------------------------------------------
__global__ void qnat_norm(const float* __restrict__ logits_ws,
                          const float* __restrict__ accum,
                          const float* __restrict__ gamma,
                          const float* __restrict__ beta,
                          float* __restrict__ out,
                          int total, float invB)
{
    const int i = blockIdx.x * blockDim.x + threadIdx.x;
    if (i >= total) return;
    const int n   = i & 3;
    const float mu  = accum[n * 2 + 0] * invB;
    const float var = accum[n * 2 + 1] * invB - mu * mu;
    const float rs  = rsqrtf(var + 1e-5f);
    out[i] = gamma[n] * (logits_ws[i] - mu) * rs + beta[n];
}

// ---------------------------------------------------------------------------
extern "C" void kernel_launch(void* const* d_in, const int* in_sizes, int n_in,
                              void* d_out, int out_size, void* d_ws, size_t ws_size,
                              hipStream_t stream)
{
    const float* x     = (const float*)d_in[0];
    const float* qp    = (const float*)d_in[1];
    const float* W1    = (const float*)d_in[2];
    const float* b1    = (const float*)d_in[3];
    const float* W2    = (const float*)d_in[4];
    const float* b2    = (const float*)d_in[5];
    const float* gamma = (const float*)d_in[6];
    const float* beta  = (const float*)d_in[7];

    const int B = in_sizes[0] / 1296;           // 65536
    float* logits_ws = (float*)d_ws;            // B*4 floats
    float* accum     = logits_ws + (size_t)B * 4;

    qnat_init<<<1, 32, 0, stream>>>(accum);

    qnat_main<<<B / 64, 128, 0, stream>>>(x, qp, W1, b1, W2, b2,
                                          logits_ws, accum);

    const int total = B * 4;
    qnat_norm<<<(total + 255) / 256, 256, 0, stream>>>(
        logits_ws, accum, gamma, beta, (float*)d_out, total, 1.0f / (float)B);
}